// RotaryEmbedding_41034117546521
// MI455X (gfx1250) — compile-verified
//
#include <hip/hip_runtime.h>
#include <hip/hip_bf16.h>

#define ROPE_H      32      // heads per token row (E / ROTARY_DIM)
#define ROPE_D      128     // rotary dim
#define ROPE_HALF   64      // rotary dim / 2
#define ROPE_E      4096    // embedding dim per token
#define THREADS     256

// -log2(10000)/64 : inv_freq[j] = exp2f(j * NEG_LOG2_BASE_OVER_HALF)
#define NEG_LOG2_BASE_OVER_HALF (-0.2076205059304601f)  // -13.287712379549448/64

struct __align__(16) f4 { float x, y, z, w; };

__global__ __launch_bounds__(THREADS) void RotaryEmbedding_41034117546521_kernel(
    const int*   __restrict__ positions,   // [B*S]
    const float* __restrict__ q,           // [B*S, 4096]
    const float* __restrict__ k,           // [B*S, 4096]
    float*       __restrict__ q_out,       // [B*S, 4096]
    float*       __restrict__ k_out)       // [B*S, 4096]
{
    __shared__ __align__(16) float qbuf[ROPE_E];   // 16 KB
    __shared__ __align__(16) float kbuf[ROPE_E];   // 16 KB
    __shared__ __align__(16) float cosb[ROPE_HALF];
    __shared__ __align__(16) float sinb[ROPE_HALF];

    const int    tid     = threadIdx.x;
    const int    row     = blockIdx.x;               // token index b*S+s
    const size_t rowBase = (size_t)row * ROPE_E;
    const float* qrow = q + rowBase;
    const float* krow = k + rowBase;

    // LDS byte offsets of the staging buffers (generic ptr low 32 bits == LDS offset)
    const unsigned qbase = (unsigned)(size_t)(void*)qbuf;
    const unsigned kbase = (unsigned)(size_t)(void*)kbuf;

    // ---- 1) async DMA the whole token row (q and k) into LDS --------------
    // Each thread issues 8 x b128 async transfers (32 KB total per block).
    // GVS mode: global address is the SGPR base plus the VGPR offset; the first VGPR gives the LDS address.
#pragma unroll
    for (int t = 0; t < 4; ++t) {
        unsigned off = (unsigned)(tid + t * THREADS) * 16u;
        asm volatile("global_load_async_to_lds_b128 %0, %1, %2"
                     :: "v"(qbase + off), "v"(off), "s"(qrow)
                     : "memory");
    }
#pragma unroll
    for (int t = 0; t < 4; ++t) {
        unsigned off = (unsigned)(tid + t * THREADS) * 16u;
        asm volatile("global_load_async_to_lds_b128 %0, %1, %2"
                     :: "v"(kbase + off), "v"(off), "s"(krow)
                     : "memory");
    }

    // ---- 2) overlap: 64 unique cos/sin values for this token --------------
    if (tid < ROPE_HALF) {
        const float pos   = (float)positions[row];
        const float theta = pos * exp2f((float)tid * NEG_LOG2_BASE_OVER_HALF);
        float s, c;
        __sincosf(theta, &s, &c);
        cosb[tid] = c;
        sinb[tid] = s;
    }

    // ---- 3) wait for async DMA, then block-wide barrier -------------------
    asm volatile("s_wait_asynccnt 0" ::: "memory");
    __syncthreads();

    // ---- 4) rotate: 1024 chunk-pair tasks (512 q + 512 k), 4 per thread ---
    // task idx in [0,512): head = idx>>4, chunk c = idx&15 covers dims [4c,4c+4)
    // out[d]      = x[d]   * cos[d] - x[d+64] * sin[d]
    // out[d+64]   = x[d+64]* cos[d] + x[d]    * sin[d]
    const f4* cos4 = (const f4*)cosb;
    const f4* sin4 = (const f4*)sinb;
    float* oq = q_out + rowBase;
    float* ok = k_out + rowBase;

#pragma unroll
    for (int t = 0; t < 4; ++t) {
        const int task   = t * THREADS + tid;        // 0..1023
        const bool is_q  = (task < 512);             // const-folded per unrolled t
        const float* buf = is_q ? qbuf : kbuf;
        float* out       = is_q ? oq : ok;
        const int idx    = task & 511;
        const int head   = idx >> 4;
        const int c      = idx & 15;
        const int baseF  = head * ROPE_D + c * 4;

        f4 x1 = *(const f4*)(buf + baseF);
        f4 x2 = *(const f4*)(buf + baseF + ROPE_HALF);
        f4 cs = cos4[c];
        f4 sn = sin4[c];

        f4 o1, o2;
        o1.x = x1.x * cs.x - x2.x * sn.x;
        o1.y = x1.y * cs.y - x2.y * sn.y;
        o1.z = x1.z * cs.z - x2.z * sn.z;
        o1.w = x1.w * cs.w - x2.w * sn.w;
        o2.x = x2.x * cs.x + x1.x * sn.x;
        o2.y = x2.y * cs.y + x1.y * sn.y;
        o2.z = x2.z * cs.z + x1.z * sn.z;
        o2.w = x2.w * cs.w + x1.w * sn.w;

        *(f4*)(out + baseF)             = o1;
        *(f4*)(out + baseF + ROPE_HALF) = o2;
    }
}

extern "C" void kernel_launch(void* const* d_in, const int* in_sizes, int n_in,
                              void* d_out, int out_size, void* d_ws, size_t ws_size,
                              hipStream_t stream) {
    const int*   positions = (const int*)d_in[0];
    const float* q         = (const float*)d_in[1];
    const float* k         = (const float*)d_in[2];
    const size_t qElems    = (size_t)in_sizes[1];    // B*S*E
    const int    rows      = (int)(qElems / ROPE_E); // B*S

    float* q_out = (float*)d_out;
    float* k_out = (float*)d_out + qElems;

    RotaryEmbedding_41034117546521_kernel<<<rows, THREADS, 0, stream>>>(
        positions, q, k, q_out, k_out);
}